// HierarchicalLoss_88064009437334
// MI455X (gfx1250) — compile-verified
//
#include <hip/hip_runtime.h>

// ---------------------------------------------------------------------------
// HierarchicalLoss for MI455X (gfx1250, wave32).
// Memory-bound kernel: 126 MB streamed once (~5.4 us at 23.3 TB/s HBM).
// The 15x15 genre-distance min is collapsed to closed-form bitmask/popcount
// logic (see analysis); WMMA f32 16x16x4 is used as the cross-lane reduction
// primitive (the only place matrix hardware is applicable in this problem).
// ---------------------------------------------------------------------------

typedef float v2f __attribute__((ext_vector_type(2)));
typedef float v8f __attribute__((ext_vector_type(8)));

#define NG 15
#define ABITS  0x001Cu   // ids {2,3,4}   : speculative-fiction children (depth 3)
#define BBITS  0x7FC0u   // ids {6..14}   : novel children (depth 3)
#define D2BITS 0x0022u   // ids {1,5}     : spec-fic, novel (depth 2)
#define RSQRT3 0.57735026918962576451f

// Sum x over all 32 lanes of the wave; every lane returns the total.
// A-matrix (16x4 f32): lane l<16 holds {A[l][0],A[l][1]}, lane l>=16 holds
// {A[l-16][2],A[l-16][3]}. With a = {x, 0} and B = all-ones, D[m][n] =
// x[m] + x[m+16]. D regs r=0..7 in lane l<16 hold D[r][l]; in lane l>=16 hold
// D[r+8][l-16]. Summing the 8 D regs gives half-sums per lane-half; one
// shfl_xor(16) completes the full 32-lane sum.
__device__ __forceinline__ float wave_sum32_wmma(float x) {
  v2f a; a.x = x;    a.y = 0.0f;
  v2f b; b.x = 1.0f; b.y = 1.0f;
  v8f c = {};
  v8f d = __builtin_amdgcn_wmma_f32_16x16x4_f32(
      /*neg_a=*/false, a, /*neg_b=*/false, b,
      /*c_mod=*/(short)0, c, /*reuse_a=*/false, /*reuse_b=*/false);
  float s = ((d[0] + d[1]) + (d[2] + d[3])) + ((d[4] + d[5]) + (d[6] + d[7]));
  s += __shfl_xor(s, 16, 32);
  return s;
}

// Deterministic block sum (256 threads = 8 waves). Result valid in thread 0.
__device__ __forceinline__ float block_sum256(float v, float* lds) {
  float w = wave_sum32_wmma(v);
  const int lane = threadIdx.x & 31;
  const int wid  = threadIdx.x >> 5;
  if (lane == 0) lds[wid] = w;
  __syncthreads();
  float total = 0.0f;
  if (threadIdx.x == 0) {
    #pragma unroll
    for (int i = 0; i < 8; ++i) total += lds[i];
  }
  return total;
}

__global__ void __launch_bounds__(256)
hier_main(const float* __restrict__ P, const float* __restrict__ T,
          float* __restrict__ partials, int B) {
  __shared__ float lds[8];
  const int ROWS = 4;
  const int base = blockIdx.x * (256 * ROWS) + threadIdx.x;

  float acc = 0.0f;
  #pragma unroll
  for (int k = 0; k < ROWS; ++k) {
    const int row = base + k * 256;
    if (row < B) {
      const float* __restrict__ p = P + row * NG;
      const float* __restrict__ t = T + row * NG;

      float bce = 0.0f;
      unsigned pos = 0u, pm = 0u;
      #pragma unroll
      for (int j = 0; j < NG; ++j) {
        const float pj = p[j];
        const float tj = t[j];
        // stable BCE-with-logits: max(p,0) - p*t + log(1 + exp(-|p|))
        const float e = __expf(-fabsf(pj));
        bce += fmaxf(pj, 0.0f) - pj * tj + __logf(1.0f + e);
        pos |= (tj != 0.0f ? 1u : 0u) << j;
        pm  |= (pj >= 0.5f ? 1u : 0u) << j;
      }

      // Closed-form min-distance-to-most-specific-targets penalty.
      const unsigned SA = pos & ABITS;
      const unsigned SB = pos & BBITS;
      const int pm0 = (int)(pm & 1u);
      const int pm1 = (int)((pm >> 1) & 1u);
      const int pm5 = (int)((pm >> 5) & 1u);
      int extra;
      if (SA | SB) {               // specifics at depth 3 (the ~always case)
        const bool hA = (SA != 0u), hB = (SB != 0u);
        extra = 2 * pm0
              + (hA ? 1 : 3) * pm1
              + (hB ? 1 : 3) * pm5
              + (hA ? 2 : 4) * __popc(pm & ABITS & ~SA)
              + (hB ? 2 : 4) * __popc(pm & BBITS & ~SB);
      } else if (pos & D2BITS) {   // specifics at depth 2: subset of {1,5}
        const int h1 = (int)((pos >> 1) & 1u);
        const int h5 = (int)((pos >> 5) & 1u);
        extra = pm0
              + (h1 ? 0 : 2) * pm1
              + (h5 ? 0 : 2) * pm5
              + (h1 ? 1 : 3) * __popc(pm & ABITS)
              + (h5 ? 1 : 3) * __popc(pm & BBITS);
      } else {                     // only 'fiction' positive: min_d = depth-1
        extra = pm1 + pm5 + 2 * (__popc(pm & ABITS) + __popc(pm & BBITS));
      }
      acc += bce + (float)extra * RSQRT3;
    }
  }

  const float total = block_sum256(acc, lds);
  if (threadIdx.x == 0) partials[blockIdx.x] = total;
}

__global__ void __launch_bounds__(256)
hier_final(const float* __restrict__ partials, int n,
           float* __restrict__ out, float scale) {
  __shared__ float lds[8];
  float acc = 0.0f;
  for (int i = threadIdx.x; i < n; i += 256) acc += partials[i];
  const float total = block_sum256(acc, lds);
  if (threadIdx.x == 0) out[0] = total * scale;
}

extern "C" void kernel_launch(void* const* d_in, const int* in_sizes, int n_in,
                              void* d_out, int out_size, void* d_ws, size_t ws_size,
                              hipStream_t stream) {
  const float* P = (const float*)d_in[0];  // predictions [B,15] f32
  const float* T = (const float*)d_in[1];  // targets     [B,15] f32
  const int B = in_sizes[0] / NG;

  const int BLOCK = 256, ROWS = 4;
  const int rowsPerBlock = BLOCK * ROWS;
  const int nblocks = (B + rowsPerBlock - 1) / rowsPerBlock;   // 1024 for B=2^20

  float* partials = (float*)d_ws;  // nblocks * 4 bytes (4 KB)

  hier_main<<<nblocks, BLOCK, 0, stream>>>(P, T, partials, B);

  const float scale = 1.0f / (15.0f * (float)B);  // extra already carries 1/sqrt(3)
  hier_final<<<1, BLOCK, 0, stream>>>(partials, nblocks, (float*)d_out, scale);
}